// CRF_4501125726705
// MI455X (gfx1250) — compile-verified
//
#include <hip/hip_runtime.h>
#include <stdint.h>

#define BSZ   64
#define SLEN  512
#define TAGS  32
#define CH    64            // rows per TDM chunk (64 rows * 128 B = 8 KB)
#define NCH   (SLEN / CH)   // 8 chunks

typedef unsigned int u32x4 __attribute__((ext_vector_type(4)));
typedef unsigned int u32x8 __attribute__((ext_vector_type(8)));

__device__ __forceinline__ float readlane_f(float v, int lane) {
    return __builtin_bit_cast(float,
        __builtin_amdgcn_readlane(__builtin_bit_cast(int, v), lane));
}

// Issue one TDM descriptor: copy a CH x TAGS f32 tile (row-major, contiguous)
// from global memory into LDS at byte offset `lds_byte`.
// D# layout per CDNA5 ISA (2D tensor, groups 0+1 only).
__device__ __forceinline__ void tdm_load_chunk(const float* gsrc, uint32_t lds_byte) {
    uint64_t ga = (uint64_t)(uintptr_t)gsrc;
    u32x4 g0;
    g0.x = 1u;                                                  // count=1 (valid), user mode
    g0.y = lds_byte;                                            // lds_addr [63:32]
    g0.z = (uint32_t)ga;                                        // global_addr[31:0]
    g0.w = ((uint32_t)(ga >> 32) & 0x01FFFFFFu) | (2u << 30);   // global_addr[56:32] | type=2
    u32x8 g1;
    g1.s0 = (2u << 16);              // workgroup_mask=0 (not in cluster), data_size=2 (4B)
    g1.s1 = ((uint32_t)TAGS << 16);  // atomic_barrier_addr=0 | tensor_dim0[15:0]=32
    g1.s2 = ((uint32_t)CH   << 16);  // tensor_dim0[31:16]=0  | tensor_dim1[15:0]=64
    g1.s3 = ((uint32_t)TAGS << 16);  // tensor_dim1[31:16]=0  | tile_dim0=32
    g1.s4 = (uint32_t)CH;            // tile_dim1=64 | tile_dim2=0
    g1.s5 = (uint32_t)TAGS;          // tensor_dim0_stride=32 (elements)
    g1.s6 = 0u;                      // stride0 hi | tensor_dim1_stride lo (unused, 2D)
    g1.s7 = 0u;
    asm volatile("tensor_load_to_lds %0, %1" :: "s"(g0), "s"(g1) : "memory");
}

__global__ void __launch_bounds__(32)
crf_viterbi_kernel(const float* __restrict__ feats,   // (B, S, TAGS) f32
                   const int*   __restrict__ mask,    // (B, S) int (prefix-true)
                   const float* __restrict__ trans,   // (TAGS, TAGS) f32
                   float*       __restrict__ out)     // (B, S) label ids as f32
{
    __shared__ float         fbuf[2][CH * TAGS];      // 16 KB feature double buffer
    __shared__ unsigned char bp[SLEN * TAGS];         // 16 KB backpointers
    __shared__ unsigned char dec[SLEN];               // decoded path (bytes)

    const int lane = threadIdx.x;                     // tag id j, wave32
    const int b    = blockIdx.x;                      // one wave per sequence
    const float* fb_base = feats + (size_t)b * SLEN * TAGS;

    // ---- sequence length L = sum(mask[b, :]) : lane-parallel + butterfly ----
    int cnt = 0;
    #pragma unroll
    for (int k = 0; k < SLEN / 32; ++k)
        cnt += (mask[b * SLEN + k * 32 + lane] != 0) ? 1 : 0;
    #pragma unroll
    for (int off = 16; off > 0; off >>= 1)
        cnt += __shfl_xor(cnt, off, 32);
    const int L = cnt;                                // 1..512, uniform

    // ---- lane j holds transition column trans[i][j] for all i in VGPRs ----
    float tc[TAGS];
    #pragma unroll
    for (int i = 0; i < TAGS; ++i) tc[i] = trans[i * TAGS + lane];
    const float tstart = trans[(TAGS - 2) * TAGS + lane];   // START row

    // ---- kick off first async tensor chunk ----
    tdm_load_chunk(fb_base, (uint32_t)(uintptr_t)&fbuf[0][0]);

    float part = 0.0f, last = 0.0f;

    for (int c = 0; c < NCH; ++c) {
        if (c + 1 < NCH) {
            tdm_load_chunk(fb_base + (size_t)(c + 1) * CH * TAGS,
                           (uint32_t)(uintptr_t)&fbuf[(c + 1) & 1][0]);
            __builtin_amdgcn_s_wait_tensorcnt(1);     // oldest (chunk c) complete
        } else {
            __builtin_amdgcn_s_wait_tensorcnt(0);
        }
        const float* fb = &fbuf[c & 1][0];

        #pragma unroll 1
        for (int r = 0; r < CH; ++r) {
            const int t = c * CH + r;                 // uniform
            const float feat = fb[r * TAGS + lane];   // ds_load_b32
            if (t == 0) {
                part = feat + tstart;                 // part0
                last = part;
                continue;
            }
            // Candidates: cand[i] = part[i] + trans[i][lane]  (independent adds)
            float bv[TAGS];
            int   bi[TAGS];
            #pragma unroll
            for (int i = 0; i < TAGS; ++i) {
                bv[i] = readlane_f(part, i) + tc[i];
                bi[i] = i;
            }
            // Binary-tree first-max argmax: left operand always has the
            // smaller index, replace only on strict '>', so ties keep the
            // lowest index (matches jnp.argmax).  Critical path: 5 levels.
            #pragma unroll
            for (int s = 1; s < TAGS; s <<= 1) {
                #pragma unroll
                for (int i = 0; i < TAGS; i += 2 * s) {
                    if (bv[i + s] > bv[i]) { bv[i] = bv[i + s]; bi[i] = bi[i + s]; }
                }
            }
            part = bv[0] + feat;                      // carried unconditionally (ref scan)
            bp[t * TAGS + lane] = (unsigned char)((t < L) ? bi[0] : 0);
            if (t <= L - 1) last = part;              // snapshot part_hist[L-1]
        }
    }

    // ---- pointer = argmax_i(last[i] + trans[i][END]), first-index tie-break ----
    float lv = last + trans[lane * TAGS + (TAGS - 1)];
    int   li = lane;
    #pragma unroll
    for (int off = 16; off > 0; off >>= 1) {
        const float ov = __shfl_xor(lv, off, 32);
        const int   oi = __shfl_xor(li, off, 32);
        if (ov > lv || (ov == lv && oi < li)) { lv = ov; li = oi; }
    }
    const int pointer = li;                           // uniform across wave

    __syncthreads();                                  // fence bp stores before lane-0 reads

    // ---- backtrace (lane 0, inherently serial): path -> LDS byte array ----
    if (lane == 0) {
        int ptr = pointer;
        dec[SLEN - 1] = (unsigned char)pointer;
        for (int k = SLEN - 2; k >= 0; --k) {
            // back[k] rows: k==L-1 -> broadcast(pointer); k>=L -> 0 (stored as 0);
            // k<L-1 -> bp at time k+1.
            const int v = (k == L - 1) ? pointer : (int)bp[(k + 1) * TAGS + ptr];
            dec[k] = (unsigned char)v;
            ptr = v;
        }
    }

    __syncthreads();                                  // fence dec before wave-wide read

    // ---- coalesced output: 16 x global_store_b32 per sequence ----
    float* ob = out + (size_t)b * SLEN;
    #pragma unroll
    for (int g = 0; g < SLEN / 32; ++g) {
        const int k = g * 32 + lane;
        ob[k] = (float)dec[k];
    }
}

extern "C" void kernel_launch(void* const* d_in, const int* in_sizes, int n_in,
                              void* d_out, int out_size, void* d_ws, size_t ws_size,
                              hipStream_t stream) {
    (void)in_sizes; (void)n_in; (void)out_size; (void)d_ws; (void)ws_size;
    const float* feats = (const float*)d_in[0];   // (64, 512, 32) f32
    const int*   mask  = (const int*)d_in[1];     // (64, 512) int
    const float* trans = (const float*)d_in[2];   // (32, 32) f32
    float*       out   = (float*)d_out;           // (64, 512) ids as f32
    crf_viterbi_kernel<<<BSZ, 32, 0, stream>>>(feats, mask, trans, out);
}